// StealNMSLoss_21603685499336
// MI455X (gfx1250) — compile-verified
//
#include <hip/hip_runtime.h>
#include <hip/hip_bf16.h>
#include <stdint.h>

typedef __attribute__((ext_vector_type(2))) float v2f;
typedef __attribute__((ext_vector_type(8))) float v8f;

#define EPSF 1.1920929e-07f
#define IMH 1024
#define IMW 1024
#define TX 32
#define TY 8

__device__ __forceinline__ uint32_t lds_byte_off(const void* p) {
  // LDS aperture: addr[31:0] of a generic pointer to __shared__ is the LDS byte offset
  return (uint32_t)(uintptr_t)p;
}

__device__ __forceinline__ void async_ld_b32(uint32_t lds_off, int gbyte_off, const float* base) {
  // gfx1250 async global->LDS DMA, GVS mode: reads global saddr + vaddr, writes the LDS offset held in vdst
  asm volatile("global_load_async_to_lds_b32 %0, %1, %2"
               :: "v"(lds_off), "v"(gbyte_off), "s"(base)
               : "memory");
}

// One-instruction XOR butterfly step via ds_swizzle (group-of-32 mode:
// and=0x1F, or=0, xor=m  ->  imm = (m<<10) | 0x1F)
__device__ __forceinline__ float wave32_sum(float x) {
  x += __int_as_float(__builtin_amdgcn_ds_swizzle(__float_as_int(x), 0x401F)); // xor 16
  x += __int_as_float(__builtin_amdgcn_ds_swizzle(__float_as_int(x), 0x201F)); // xor 8
  x += __int_as_float(__builtin_amdgcn_ds_swizzle(__float_as_int(x), 0x101F)); // xor 4
  x += __int_as_float(__builtin_amdgcn_ds_swizzle(__float_as_int(x), 0x081F)); // xor 2
  x += __int_as_float(__builtin_amdgcn_ds_swizzle(__float_as_int(x), 0x041F)); // xor 1
  return x;
}

// Exact two-stage Sobel with replicate clamping at BOTH stages (border blocks).
__device__ __forceinline__ float compute_v_border(const float (&te)[TY + 4][TX + 4],
                                                  int ly, int lx, int gy0, int gx0) {
  const int oy0 = (gy0 == 0) ? 0 : -1;
  const int oy2 = (gy0 == IMH - 1) ? 0 : 1;
  const int ox0 = (gx0 == 0) ? 0 : -1;
  const int ox2 = (gx0 == IMW - 1) ? 0 : 1;
  auto TE = [&](int a, int b) { return te[ly + a][lx + b]; };
  auto GX = [&](int a, int b) {
    return 0.125f * ((TE(a - 1, b + 1) - TE(a - 1, b - 1)) +
                     2.0f * (TE(a, b + 1) - TE(a, b - 1)) +
                     (TE(a + 1, b + 1) - TE(a + 1, b - 1)));
  };
  auto GY = [&](int a, int b) {
    return 0.125f * ((TE(a + 1, b - 1) - TE(a - 1, b - 1)) +
                     2.0f * (TE(a + 1, b) - TE(a - 1, b)) +
                     (TE(a + 1, b + 1) - TE(a - 1, b + 1)));
  };
  float gxx = 0.125f * ((GX(oy0, ox2) - GX(oy0, ox0)) +
                        2.0f * (GX(0, ox2) - GX(0, ox0)) +
                        (GX(oy2, ox2) - GX(oy2, ox0)));
  float gxy = 0.125f * ((GY(oy0, ox2) - GY(oy0, ox0)) +
                        2.0f * (GY(0, ox2) - GY(0, ox0)) +
                        (GY(oy2, ox2) - GY(oy2, ox0)));
  float gyy = 0.125f * ((GY(oy2, ox0) - GY(oy0, ox0)) +
                        2.0f * (GY(oy2, 0) - GY(oy0, 0)) +
                        (GY(oy2, ox2) - GY(oy0, ox2)));
  float t = EPSF - gxy;
  float sg = (t > 0.0f) ? 1.0f : ((t < 0.0f) ? -1.0f : 0.0f);
  return gyy * sg / (gxx + EPSF);
}

__global__ __launch_bounds__(256) void nms_loss_kernel(const float* __restrict__ tl,
                                                       const float* __restrict__ pl,
                                                       float* __restrict__ out) {
  __shared__ float te[TY + 4][TX + 4];  // true_labels tile, halo 2 (replicate)
  __shared__ float pe[TY + 2][TX + 2];  // exp(pred) tile, halo 1 (reflect)
  __shared__ float va[TY][TX + 4];      // vertical [1,4,6,4,1]   (for gxx)
  __shared__ float vb[TY][TX + 4];      // vertical [1,0,-2,0,1]  (for gyy)
  __shared__ float vc[TY][TX + 4];      // vertical [-1,-2,0,2,1] (for gxy)
  __shared__ float red[256];

  const int tid = threadIdx.y * TX + threadIdx.x;
  const int x0 = blockIdx.x * TX;
  const int y0 = blockIdx.y * TY;
  const size_t plane = (size_t)blockIdx.z * (size_t)(IMH * IMW);
  const float* tlp = tl + plane;
  const float* plp = pl + plane;

  // --- true_labels halo tile via async global->LDS DMA (replicate clamp) ---
  const uint32_t te_base = lds_byte_off(&te[0][0]);
  for (int i = tid; i < (TY + 4) * (TX + 4); i += 256) {
    int r = i / (TX + 4), c = i - r * (TX + 4);
    int gy = min(max(y0 - 2 + r, 0), IMH - 1);
    int gx = min(max(x0 - 2 + c, 0), IMW - 1);
    async_ld_b32(te_base + (uint32_t)i * 4u, (gy * IMW + gx) * 4, tlp);
  }

  // --- exp(pred) halo tile (reflect pad), exp computed once per element ---
  for (int i = tid; i < (TY + 2) * (TX + 2); i += 256) {
    int r = i / (TX + 2), c = i - r * (TX + 2);
    int gy = y0 - 1 + r;
    gy = gy < 0 ? -gy : (gy > IMH - 1 ? 2 * (IMH - 1) - gy : gy);
    int gx = x0 - 1 + c;
    gx = gx < 0 ? -gx : (gx > IMW - 1 ? 2 * (IMW - 1) - gx : gx);
    pe[r][c] = __expf(plp[gy * IMW + gx]);
  }

  asm volatile("s_wait_asynccnt 0" ::: "memory");
  __syncthreads();

  const bool border = (blockIdx.x == 0) | (blockIdx.x == gridDim.x - 1) |
                      (blockIdx.y == 0) | (blockIdx.y == gridDim.y - 1);

  float v;
  if (border) {
    const int lx = threadIdx.x + 2, ly = threadIdx.y + 2;
    v = compute_v_border(te, ly, lx, y0 + threadIdx.y, x0 + threadIdx.x);
  } else {
    // Stage 1: shared vertical sums of the separable composite 5x5 kernels.
    // One position per (output row, tile column incl. halo2): TY x (TX+4).
    for (int i = tid; i < TY * (TX + 4); i += 256) {
      int r = i / (TX + 4), c = i - r * (TX + 4);
      float t0 = te[r][c], t1 = te[r + 1][c], t2 = te[r + 2][c];
      float t3 = te[r + 3][c], t4 = te[r + 4][c];
      va[r][c] = t0 + 4.0f * t1 + 6.0f * t2 + 4.0f * t3 + t4;
      vb[r][c] = t0 - 2.0f * t2 + t4;
      vc[r][c] = (t4 - t0) + 2.0f * (t3 - t1);
    }
    __syncthreads();

    // Stage 2: cheap horizontal combines.
    const int r = threadIdx.y, cx = threadIdx.x + 2;
    const float inv64 = 1.0f / 64.0f;
    float gxx = inv64 * (va[r][cx - 2] - 2.0f * va[r][cx] + va[r][cx + 2]);
    float gyy = inv64 * (vb[r][cx - 2] + 4.0f * vb[r][cx - 1] + 6.0f * vb[r][cx] +
                         4.0f * vb[r][cx + 1] + vb[r][cx + 2]);
    float gxy = inv64 * ((vc[r][cx + 2] - vc[r][cx - 2]) +
                         2.0f * (vc[r][cx + 1] - vc[r][cx - 1]));
    float t = EPSF - gxy;
    float sg = (t > 0.0f) ? 1.0f : ((t < 0.0f) ? -1.0f : 0.0f);
    v = gyy * sg / (gxx + EPSF);
  }

  const int px = threadIdx.x + 1, py = threadIdx.y + 1;
  float e0 = pe[py][px];
  float hsum = pe[py][px - 1] + e0 + pe[py][px + 1];
  float vsum = pe[py - 1][px] + e0 + pe[py + 1][px];
  float lsum = pe[py - 1][px - 1] + e0 + pe[py + 1][px + 1];
  float csum = pe[py - 1][px + 1] + e0 + pe[py + 1][px - 1];

  // round(atan(v)*5/pi) bins via tangent thresholds (NaN -> no contribution)
  const float T1 = 0.32491969623290634f;  // tan(pi/10)
  const float T3 = 1.37638192047117353f;  // tan(3pi/10)
  float val = 0.0f;
  if (v >= -T1 && v < T1)       val = e0 / (hsum + EPSF);  // bin 0: horiz
  else if (v >= T1 && v < T3)   val = e0 / (csum + EPSF);  // bin 1: anti-diag
  else if (v >= T3)             val = e0 / (vsum + EPSF);  // bin 2: vert
  else if (v < -T3)             val = e0 / (lsum + EPSF);  // bin 3: diag
  // bin 4 and NaN contribute 0

  // --- block reduction: 256 -> 64 -> V_WMMA_F32_16X16X4_F32 with A = ones ---
  red[tid] = val;
  __syncthreads();
  if (tid < 32) {  // wave 0, EXEC all ones
    v2f b;
    b.x = red[tid] + red[tid + 64] + red[tid + 128] + red[tid + 192];
    b.y = red[tid + 32] + red[tid + 96] + red[tid + 160] + red[tid + 224];
    v2f a;
    a.x = 1.0f; a.y = 1.0f;
    v8f c = {};
    // D = ones(16x4) * B(4x16): every row of D holds the column sums of B, so
    // the full D matrix sums to 16 * (sum of all 64 B elements) for ANY layout.
    c = __builtin_amdgcn_wmma_f32_16x16x4_f32(false, a, false, b, (short)0, c,
                                              false, false);
    float t = c[0] + c[1] + c[2] + c[3] + c[4] + c[5] + c[6] + c[7];
    t = wave32_sum(t);
    if (tid == 0) atomicAdd(out, t * (1.0f / 16.0f));
  }
}

extern "C" void kernel_launch(void* const* d_in, const int* in_sizes, int n_in,
                              void* d_out, int out_size, void* d_ws, size_t ws_size,
                              hipStream_t stream) {
  const float* tl = (const float*)d_in[0];   // true_labels [8,2,1024,1024] f32
  const float* pl = (const float*)d_in[1];   // pred_labels [8,2,1024,1024] f32
  float* out = (float*)d_out;                // scalar f32
  (void)in_sizes; (void)n_in; (void)out_size; (void)d_ws; (void)ws_size;

  hipMemsetAsync(out, 0, sizeof(float), stream);  // graph-capturable memset node

  dim3 grid(IMW / TX, IMH / TY, 16);  // 16 = B*C planes
  dim3 block(TX, TY);
  nms_loss_kernel<<<grid, block, 0, stream>>>(tl, pl, out);
}